// DropInFFM_40338332844219
// MI455X (gfx1250) — compile-verified
//
#include <hip/hip_runtime.h>
#include <hip/hip_bf16.h>
#include <math.h>
#include <stdint.h>

// ---------------- problem constants (from reference) ----------------
#define Mdim   32
#define Cdim   8
#define OUTF   128
#define INF    128
#define Tn     2048
#define Bsz    32
#define SRW    (2*Mdim*Cdim)       // 512
#define NROWS  (Tn*Bsz)            // 65536
#define EPSF   1e-5f

typedef float v2f __attribute__((ext_vector_type(2)));
typedef float v8f __attribute__((ext_vector_type(8)));

__device__ __forceinline__ float sigmoidf_(float x) {
    return 1.0f / (1.0f + __expf(-x));
}

#define WMMA_F32(a, b, c) \
    __builtin_amdgcn_wmma_f32_16x16x4_f32(false, (a), false, (b), (short)0, (c), false, false)

// =====================================================================
// Kernel 1: xg[65536 x 32] = (X@Wp^T[:, :32]) * sigmoid(X@Wp^T[:,160:192])
// One wave32 computes one 16x16 y-tile AND the matching in_gate tile
// (shared A loads, two V_WMMA_F32_16X16X4_F32 accumulators), then fuses
// the gate and writes compact xg. thru/out_gate are NOT materialized --
// kernel 3 recomputes them (flops are free; HBM bytes are not).
// =====================================================================
__global__ __launch_bounds__(64)
void ffm_xg_gemm(const float* __restrict__ X,
                 const float* __restrict__ Wp,
                 const float* __restrict__ bp,
                 float* __restrict__ xg)
{
    const int lane = threadIdx.x;        // 0..31 (wave32)
    const int l    = lane & 15;
    const int hi   = lane >> 4;          // K sub-pair selector
    const int row0 = blockIdx.x * 16;
    const int m0   = threadIdx.y * 16;   // 0 or 16

    const float* arow = X  + (size_t)(row0 + l) * INF + 2 * hi;
    const float* brY  = Wp + (size_t)(m0 + l) * INF + 2 * hi;          // y rows
    const float* brG  = Wp + (size_t)(160 + m0 + l) * INF + 2 * hi;    // in_gate rows

    v8f accY = {}, accG = {};
#pragma unroll 4
    for (int k0 = 0; k0 < INF; k0 += 4) {
        v2f a  = *(const v2f*)(arow + k0);
        v2f by = *(const v2f*)(brY + k0);
        v2f bg = *(const v2f*)(brG + k0);
        accY = WMMA_F32(a, by, accY);
        accG = WMMA_F32(a, bg, accG);
    }

    const float bY = bp[m0 + l];
    const float bG = bp[160 + m0 + l];
#pragma unroll
    for (int v = 0; v < 8; ++v) {
        const int r = v + 8 * hi;                       // C/D row layout
        xg[(size_t)(row0 + r) * Mdim + m0 + l] =
            (accY[v] + bY) * sigmoidf_(accG[v] + bG);
    }
}

// =====================================================================
// Kernel 2: complex diagonal scan  s_t = gamma*s_{t-1} + xg_t
// 8192 independent (b,m,c) sequences = 256 wave32s; one state per lane.
// Critical path per step = 2 dependent FMAs; xg loads for future t are
// independent (compiler runs ahead; global_prefetch_b8 helps the strided
// time walk). Emits SR rows in the reference's [m*16+c*2+{re,im}] layout
// and s_last (complex64 interleaved) into the output tail.
// =====================================================================
__global__ __launch_bounds__(256)
void ffm_scan(const float* __restrict__ xg,
              const float* __restrict__ a_decay,
              const float* __restrict__ b_freq,
              float* __restrict__ sr,
              float* __restrict__ s_last)
{
    const int g = blockIdx.x * 256 + threadIdx.x;   // 0..8191
    const int c = g & (Cdim - 1);
    const int m = (g >> 3) & (Mdim - 1);
    const int b = g >> 8;

    const float decay = __expf(-fabsf(a_decay[m]));
    const float gr = decay * cosf(b_freq[c]);
    const float gi = decay * sinf(b_freq[c]);

    float s_re = 0.0f, s_im = 0.0f;
    const size_t outoff = (size_t)m * (2 * Cdim) + (size_t)c * 2;
    const float* xsrc = xg + (size_t)b * Mdim + m;

    for (int t = 0; t < Tn; ++t) {
        const size_t row = (size_t)t * Bsz + b;
        if (t + 16 < Tn)
            __builtin_prefetch(xsrc + (size_t)(t + 16) * Bsz * Mdim, 0, 1);

        const float xv = xsrc[(size_t)t * Bsz * Mdim];

        const float nr = fmaf(gr, s_re, fmaf(-gi, s_im, xv));
        const float ni = fmaf(gr, s_im,  gi * s_re);
        s_re = nr; s_im = ni;

        float* dst = sr + row * SRW + outoff;
        dst[0] = s_re;
        dst[1] = s_im;
    }

    // s_last: [B, M, C] complex64 (re,im interleaved)
    const size_t li = (((size_t)b * Mdim + m) * Cdim + c) * 2;
    s_last[li + 0] = s_re;
    s_last[li + 1] = s_im;
}

// =====================================================================
// Kernel 3: per 16-row stripe: stage SR tile (16x512, 32KB) into LDS via
// CDNA5 async global->LDS (ASYNCcnt), then 8 waves compute
//   z tile   = SR_lds @ W_mix^T + b_mix      (K=512, A from ds_load_b64)
//   thru/og  = X @ W_pre^T cols 32+n0 / 192+n0  (K=128, recomputed)
// followed by LayerNorm(128) over LDS z and the gated output mix.
// =====================================================================
__global__ __launch_bounds__(256)
void ffm_mix_ln_out(const float* __restrict__ sr,
                    const float* __restrict__ Wm,
                    const float* __restrict__ bm,
                    const float* __restrict__ X,
                    const float* __restrict__ Wp,
                    const float* __restrict__ bp,
                    float* __restrict__ out)
{
    __shared__ float tileA[16 * SRW];     // 32 KB SR stripe
    __shared__ float zt[16][OUTF];        // 8 KB z tile
    __shared__ float mu_s[16], rstd_s[16];

    const int lane = threadIdx.x;
    const int l    = lane & 15;
    const int hi   = lane >> 4;
    const int w    = threadIdx.y;          // 0..7 -> column tile
    const int tid  = w * 32 + lane;        // 0..255
    const int row0 = blockIdx.x * 16;
    const int n0   = w * 16;

    // ---- async-copy SR stripe to LDS: 8192 floats = 2048 x b128 / 256 thr ----
    {
        const float* gsrc = sr + (size_t)row0 * SRW + (size_t)tid * 4;
        const uint32_t laddr = (uint32_t)(uintptr_t)tileA + (uint32_t)tid * 16u;
#pragma unroll
        for (int i = 0; i < 8; ++i) {
            asm volatile("global_load_async_to_lds_b128 %0, %1, off"
                         :: "v"(laddr + (uint32_t)i * 4096u),
                            "v"(gsrc + (size_t)i * 1024)
                         : "memory");
        }
        asm volatile("s_wait_asynccnt 0x0" ::: "memory");
    }
    __syncthreads();

    // ---- z tile: K=512, A operand broadcast from LDS ----
    const float* aL = tileA + (size_t)l * SRW + 2 * hi;
    const float* bz = Wm + (size_t)(n0 + l) * SRW + 2 * hi;
    v8f accZ = {};
#pragma unroll 8
    for (int k0 = 0; k0 < SRW; k0 += 4) {
        v2f a = *(const v2f*)(aL + k0);        // ds_load_b64
        v2f b = *(const v2f*)(bz + k0);
        accZ = WMMA_F32(a, b, accZ);
    }

    // ---- thru + out_gate tiles recomputed from X (K=128, shared A) ----
    const float* ax = X  + (size_t)(row0 + l) * INF + 2 * hi;
    const float* bt = Wp + (size_t)(32 + n0 + l) * INF + 2 * hi;
    const float* bo = Wp + (size_t)(192 + n0 + l) * INF + 2 * hi;
    v8f accT = {}, accO = {};
#pragma unroll 4
    for (int k0 = 0; k0 < INF; k0 += 4) {
        v2f a  = *(const v2f*)(ax + k0);
        v2f b1 = *(const v2f*)(bt + k0);
        v2f b2 = *(const v2f*)(bo + k0);
        accT = WMMA_F32(a, b1, accT);
        accO = WMMA_F32(a, b2, accO);
    }

    const float biasZ = bm[n0 + l];
#pragma unroll
    for (int v = 0; v < 8; ++v)
        zt[v + 8 * hi][n0 + l] = accZ[v] + biasZ;
    __syncthreads();

    // ---- LayerNorm stats: one thread per row (16 x 128, biased var) ----
    if (tid < 16) {
        float s = 0.0f, s2 = 0.0f;
#pragma unroll 8
        for (int n = 0; n < OUTF; ++n) {
            const float z = zt[tid][n];
            s += z; s2 += z * z;
        }
        const float mu  = s  * (1.0f / OUTF);
        const float var = s2 * (1.0f / OUTF) - mu * mu;
        mu_s[tid]   = mu;
        rstd_s[tid] = rsqrtf(var + EPSF);
    }
    __syncthreads();

    // ---- gated output from in-register thru/og tiles ----
    const float bT = bp[32 + n0 + l];
    const float bO = bp[192 + n0 + l];
#pragma unroll
    for (int v = 0; v < 8; ++v) {
        const int r  = v + 8 * hi;
        const float zn = (zt[r][n0 + l] - mu_s[r]) * rstd_s[r];
        const float th = accT[v] + bT;
        const float og = sigmoidf_(accO[v] + bO);
        out[(size_t)(row0 + r) * OUTF + n0 + l] = zn * og + th * (1.0f - og);
    }
}

// =====================================================================
// host-side launcher
// =====================================================================
extern "C" void kernel_launch(void* const* d_in, const int* in_sizes, int n_in,
                              void* d_out, int out_size, void* d_ws, size_t ws_size,
                              hipStream_t stream)
{
    const float* x     = (const float*)d_in[0];   // [T,B,128]
    const float* W_pre = (const float*)d_in[1];   // [320,128]
    const float* b_pre = (const float*)d_in[2];   // [320]
    const float* W_mix = (const float*)d_in[3];   // [128,512]
    const float* b_mix = (const float*)d_in[4];   // [128]
    const float* a_dec = (const float*)d_in[5];   // [32]
    const float* b_frq = (const float*)d_in[6];   // [8]

    float* out    = (float*)d_out;                        // y_out [T,B,128]
    float* s_last = out + (size_t)Tn * Bsz * OUTF;        // [B,M,C] complex64

    float* xg = (float*)d_ws;                             // 65536*32  f32 (8.4 MB)
    float* sr = xg + (size_t)NROWS * Mdim;                // 65536*512 f32 (134 MB)

    {   // xg gemm: 4096 row tiles, 2 waves/block (y + in_gate fused)
        dim3 blk(32, 2), grd(NROWS / 16);
        ffm_xg_gemm<<<grd, blk, 0, stream>>>(x, W_pre, b_pre, xg);
    }
    {   // scan: 8192 sequences
        ffm_scan<<<(8192 / 256), 256, 0, stream>>>(xg, a_dec, b_frq, sr, s_last);
    }
    {   // mix gemm + LN + gate: 4096 row stripes, 8 waves/block
        dim3 blk(32, 8), grd(NROWS / 16);
        ffm_mix_ln_out<<<grd, blk, 0, stream>>>(sr, W_mix, b_mix, x, W_pre, b_pre, out);
    }
}